// EdgeToNodeConv_73057393705162
// MI455X (gfx1250) — compile-verified
//
#include <hip/hip_runtime.h>
#include <hip/hip_bf16.h>

typedef float v2f __attribute__((ext_vector_type(2)));
typedef float v8f __attribute__((ext_vector_type(8)));

constexpr int EDGE_DIM   = 128;
constexpr int NODE_DIM   = 256;
constexpr int TILE_E     = 16;
constexpr int LDS_STRIDE = 132;   // words; 8B-aligned rows, conflict-free b64 fragment reads

// No-return fp32 atomic add in saddr form: SGPR base + 32-bit VGPR byte offset.
__device__ __forceinline__ void atomic_add_f32_saddr(float* base, unsigned byte_off, float v) {
#if defined(__gfx1250__)
    asm volatile("global_atomic_add_f32 %0, %1, %2 scope:SCOPE_DEV"
                 :: "v"(byte_off), "v"(v), "s"(base)
                 : "memory");
#else
    __hip_atomic_fetch_add(reinterpret_cast<float*>(reinterpret_cast<char*>(base) + byte_off),
                           v, __ATOMIC_RELAXED, __HIP_MEMORY_SCOPE_AGENT);
#endif
}

// Fragment-order packed weights: for k-pair index kq (k = 2*kq + j, kq in [0,64)),
//   WtP[(kq*256 + n)*2 + j] = W[n*128 + 2*kq + j]
// so a WMMA lane reads its (k, k+1) B-pair as ONE b64, contiguous across lanes.
__global__ void wt_pack_kernel(const float* __restrict__ W, float* __restrict__ WtP) {
    int idx = blockIdx.x * blockDim.x + threadIdx.x;   // 0 .. 128*256-1
    int kq  = idx >> 9;          // / 512
    int rem = idx & 511;
    int n   = rem >> 1;
    int j   = rem & 1;
    WtP[idx] = W[n * EDGE_DIM + 2 * kq + j];
}

template <bool USE_WT>
__global__ __launch_bounds__(256)
void edge2node_wmma_kernel(const float* __restrict__ A,        // [E,128] edge features
                           const int*   __restrict__ edge_idx, // [2,E]
                           const float* __restrict__ W,        // WtP packed if USE_WT else W [256,128]
                           const float* __restrict__ bias,     // [256]
                           float*       __restrict__ out,      // [n_nodes,256], pre-zeroed
                           int E) {
    __shared__ float    sA[TILE_E * LDS_STRIDE];
    __shared__ unsigned sSrcB[TILE_E];   // node-row BYTE offsets (node * 1024)
    __shared__ unsigned sDstB[TILE_E];

    const int tid = threadIdx.x;
    const int e0  = blockIdx.x * TILE_E;

    // ---- stage 16x128 A tile into LDS (coalesced, non-temporal: streamed once) ----
    {
        const int gbase = tid * 2;
        #pragma unroll
        for (int round = 0; round < 4; ++round) {
            int g   = gbase + round * 512;       // flat index into [16][128]
            int row = g >> 7;
            int col = g & 127;
            int er  = e0 + row;
            if (er >= E) er = E - 1;             // clamp (tail-safe; never scattered)
            v2f v = __builtin_nontemporal_load(
                        reinterpret_cast<const v2f*>(A + (size_t)er * EDGE_DIM + col));
            *reinterpret_cast<v2f*>(&sA[row * LDS_STRIDE + col]) = v;
        }
        if (tid < TILE_E) {
            int er = e0 + tid;
            sSrcB[tid] = (er < E) ? (unsigned)edge_idx[er] * (NODE_DIM * 4u) : 0u;
        } else if (tid < 2 * TILE_E) {
            int t  = tid - TILE_E;
            int er = e0 + t;
            sDstB[t] = (er < E) ? (unsigned)edge_idx[(size_t)E + er] * (NODE_DIM * 4u) : 0u;
        }
    }
    __syncthreads();

    const int lane = tid & 31;
    const int wave = tid >> 5;
    const int half = lane >> 4;   // K-half select (f32 WMMA A/B VGPR layout)
    const int l16  = lane & 15;

    const int n0 = wave * 32 + l16;   // 8 waves * 2 N-tiles * 16 = 256 columns
    const int n1 = n0 + 16;

    // accumulators seeded with bias (C/D layout: all 8 VGPRs share column n per lane)
    v8f acc0, acc1;
    {
        const float b0 = bias[n0];
        const float b1 = bias[n1];
        #pragma unroll
        for (int r = 0; r < 8; ++r) { acc0[r] = b0; acc1[r] = b1; }
    }

    const float* sArow = &sA[l16 * LDS_STRIDE + 2 * half];

    // ---- K loop: 32 steps of v_wmma_f32_16x16x4_f32, two independent accumulators ----
    #pragma unroll 4
    for (int kk = 0; kk < EDGE_DIM / 4; ++kk) {
        const int k0 = kk * 4;
        // A fragment: lanes 0-15 hold K=k0,k0+1 ; lanes 16-31 hold K=k0+2,k0+3
        v2f a = *reinterpret_cast<const v2f*>(sArow + k0);

        v2f b0, b1;
        if (USE_WT) {
            // packed: one b64 per fragment, lane-contiguous
            const int kq = kk * 2 + half;
            const float* wp = W + (size_t)kq * (NODE_DIM * 2);
            b0 = *reinterpret_cast<const v2f*>(wp + n0 * 2);
            b1 = *reinterpret_cast<const v2f*>(wp + n1 * 2);
        } else {
            b0 = *reinterpret_cast<const v2f*>(W + (size_t)n0 * EDGE_DIM + k0 + 2 * half);
            b1 = *reinterpret_cast<const v2f*>(W + (size_t)n1 * EDGE_DIM + k0 + 2 * half);
        }

        // D = A x B + C, exact fp32 (16x16x4)
        acc0 = __builtin_amdgcn_wmma_f32_16x16x4_f32(false, a, false, b0, (short)0, acc0, false, false);
        acc1 = __builtin_amdgcn_wmma_f32_16x16x4_f32(false, a, false, b1, (short)0, acc1, false, false);
    }

    // ---- fused scatter: saddr-form fp32 atomics into src/dst node rows (L2-resident) ----
    const unsigned n0b = (unsigned)n0 * 4u;
    const unsigned n1b = (unsigned)n1 * 4u;
    const int mbase = half * 8;
    if (e0 + TILE_E <= E) {
        // full tile: no per-row guard, EXEC stays all-ones
        #pragma unroll
        for (int r = 0; r < 8; ++r) {
            const int m = mbase + r;
            const unsigned sb = sSrcB[m];
            const unsigned db = sDstB[m];
            const float v0 = acc0[r];
            const float v1 = acc1[r];
            atomic_add_f32_saddr(out, sb + n0b, v0);
            atomic_add_f32_saddr(out, sb + n1b, v1);
            atomic_add_f32_saddr(out, db + n0b, v0);
            atomic_add_f32_saddr(out, db + n1b, v1);
        }
    } else {
        #pragma unroll
        for (int r = 0; r < 8; ++r) {
            const int m = mbase + r;
            if (e0 + m < E) {
                const unsigned sb = sSrcB[m];
                const unsigned db = sDstB[m];
                const float v0 = acc0[r];
                const float v1 = acc1[r];
                atomic_add_f32_saddr(out, sb + n0b, v0);
                atomic_add_f32_saddr(out, sb + n1b, v1);
                atomic_add_f32_saddr(out, db + n0b, v0);
                atomic_add_f32_saddr(out, db + n1b, v1);
            }
        }
    }
}

extern "C" void kernel_launch(void* const* d_in, const int* in_sizes, int n_in,
                              void* d_out, int out_size, void* d_ws, size_t ws_size,
                              hipStream_t stream) {
    const float* edge_features = (const float*)d_in[0];
    const int*   edge_index    = (const int*)  d_in[1];
    // d_in[2] = n_nodes (scalar) -- implied by out_size
    const float* W    = (const float*)d_in[3];
    const float* bias = (const float*)d_in[4];
    float* out = (float*)d_out;

    const int E = in_sizes[0] / EDGE_DIM;

    // output is poisoned by the harness; atomics require zeroed destination
    hipMemsetAsync(d_out, 0, (size_t)out_size * sizeof(float), stream);

    const int grid = (E + TILE_E - 1) / TILE_E;
    const size_t wt_bytes = (size_t)EDGE_DIM * NODE_DIM * sizeof(float);

    if (ws_size >= wt_bytes) {
        float* WtP = (float*)d_ws;
        wt_pack_kernel<<<(EDGE_DIM * NODE_DIM) / 256, 256, 0, stream>>>(W, WtP);
        edge2node_wmma_kernel<true><<<grid, 256, 0, stream>>>(edge_features, edge_index,
                                                              WtP, bias, out, E);
    } else {
        edge2node_wmma_kernel<false><<<grid, 256, 0, stream>>>(edge_features, edge_index,
                                                               W, bias, out, E);
    }
}